// BoWModel_4097398800503
// MI455X (gfx1250) — compile-verified
//
#include <hip/hip_runtime.h>

typedef float v2f __attribute__((ext_vector_type(2)));
typedef float v4f __attribute__((ext_vector_type(4)));
typedef float v8f __attribute__((ext_vector_type(8)));

constexpr int BATCH = 2048;
constexpr int SEQ   = 50;
constexpr int HID   = 256;   // hidden width
constexpr int BEO   = 512;   // beo input width
constexpr int NEMB  = 64;
constexpr int NPRED = 5;
constexpr int HS    = 260;   // padded LDS row stride in floats:
                             //  - multiple of 4 => 16B-aligned b128 stores
                             //  - (4*row + k) % 64 distinct over 16 rows => conflict-free b64 loads

__device__ __forceinline__ float lrelu(float v) { return v >= 0.f ? v : 0.2f * v; }
__device__ __forceinline__ v4f lrelu4(v4f v) {
    v4f r; r.x = lrelu(v.x); r.y = lrelu(v.y); r.z = lrelu(v.z); r.w = lrelu(v.w);
    return r;
}

// Head GEMMs on a 16x256 activation tile held in LDS (row stride HS).
//   out_e[16x64] = h @ We + be     (waves 0..3, one 16-col tile each)
//   out_p[16x5]  = h @ Wp + bp     (wave 4, cols 5..15 zero-padded)
// Uses V_WMMA_F32_16X16X4_F32, K swept 4 at a time (64 steps).
__device__ __forceinline__ void heads_wmma(
    const float* __restrict__ hsm,
    const float* __restrict__ Wp, const float* __restrict__ bp,
    const float* __restrict__ We, const float* __restrict__ be,
    float* __restrict__ out_e, float* __restrict__ out_p, int row0)
{
    const int wave = threadIdx.x >> 5;
    const int lane = threadIdx.x & 31;
    const int half = lane >> 4;   // which K-pair this lane holds
    const int lr   = lane & 15;   // M for A-fetch, N for B-fetch/stores

    if (wave < 4) {
        const int n0 = wave * 16;
        v8f c = {};
        #pragma unroll 4
        for (int kk = 0; kk < HID; kk += 4) {
            const int ka = kk + half * 2;
            v2f a = *(const v2f*)&hsm[lr * HS + ka];          // A[M=lr][ka..ka+1]
            v2f b;
            b.x = We[ka * NEMB + n0 + lr];                    // B[ka][n0+lr]
            b.y = We[(ka + 1) * NEMB + n0 + lr];
            c = __builtin_amdgcn_wmma_f32_16x16x4_f32(false, a, false, b,
                                                      (short)0, c, false, false);
        }
        const float bias = be[n0 + lr];
        #pragma unroll
        for (int i = 0; i < 8; ++i)
            out_e[(size_t)(row0 + i + 8 * half) * NEMB + n0 + lr] = c[i] + bias;
    } else if (wave == 4) {
        v8f c = {};
        const int nsafe = lr < NPRED ? lr : 0;                // keep loads in-bounds
        #pragma unroll 4
        for (int kk = 0; kk < HID; kk += 4) {
            const int ka = kk + half * 2;
            v2f a = *(const v2f*)&hsm[lr * HS + ka];
            float b0 = Wp[ka * NPRED + nsafe];
            float b1 = Wp[(ka + 1) * NPRED + nsafe];
            v2f b;
            b.x = lr < NPRED ? b0 : 0.f;                      // branchless pad -> EXEC stays full
            b.y = lr < NPRED ? b1 : 0.f;
            c = __builtin_amdgcn_wmma_f32_16x16x4_f32(false, a, false, b,
                                                      (short)0, c, false, false);
        }
        if (lr < NPRED) {
            const float bias = bp[lr];
            #pragma unroll
            for (int i = 0; i < 8; ++i)
                out_p[(size_t)(row0 + i + 8 * half) * NPRED + lr] = c[i] + bias;
        }
    }
}

__global__ __launch_bounds__(256) void bow_beo_kernel(
    const int*   __restrict__ s,   const float* __restrict__ x,
    const float* __restrict__ Wbh, const float* __restrict__ bbh,
    const float* __restrict__ Wbp, const float* __restrict__ bbp,
    const float* __restrict__ Wbe, const float* __restrict__ bbe,
    const float* __restrict__ Weh, const float* __restrict__ beh,
    const float* __restrict__ Wep, const float* __restrict__ bep,
    const float* __restrict__ Wee, const float* __restrict__ bee,
    float* __restrict__ out)
{
    __shared__ float          hsm[16 * HS];      // 16x256 activation tile, padded
    __shared__ int            stok[16 * SEQ];
    __shared__ unsigned char  sflag[16 * SEQ];

    float* bow_e = out;
    float* beo_e = out + (size_t)BATCH * NEMB;
    float* bow_p = out + (size_t)2 * BATCH * NEMB;
    float* beo_p = out + (size_t)2 * BATCH * NEMB + (size_t)BATCH * NPRED;

    const int  tid    = threadIdx.x;
    const bool is_beo = (int)blockIdx.x >= (BATCH / 16);
    const int  row0   = (is_beo ? ((int)blockIdx.x - BATCH / 16) : (int)blockIdx.x) * 16;

    if (!is_beo) {
        // ---------------- BoW path: dedup embedding-bag ----------------
        for (int idx = tid; idx < 16 * SEQ; idx += 256)
            stok[idx] = s[(size_t)row0 * SEQ + idx];
        __syncthreads();

        // first-occurrence flags (scatter .set semantics == sum over unique tokens)
        for (int idx = tid; idx < 16 * SEQ; idx += 256) {
            const int r = idx / SEQ, j = idx % SEQ;
            const int tok = stok[r * SEQ + j];
            unsigned char f = 1;
            for (int k = 0; k < j; ++k)
                if (stok[r * SEQ + k] == tok) { f = 0; break; }
            sflag[idx] = f;
        }
        __syncthreads();

        // Gather-sum, vectorized b128: 64 lanes own 4 dims each (dims 4q..4q+3);
        // four 64-thread groups each cover 4 of the 16 rows. Per token-row the
        // group reads 1KB contiguous (64 lanes x 16B) -> coalesced global_load_b128.
        // Flag branch depends only on (r,j) -> uniform per wave (rg = tid>>6).
        const int q  = tid & 63;
        const int rg = tid >> 6;
        const v4f bias4 = *(const v4f*)&bbh[4 * q];
        #pragma unroll
        for (int rr = 0; rr < 4; ++rr) {
            const int r = rg * 4 + rr;
            v4f acc = bias4;
            for (int j = 0; j < SEQ; ++j) {
                if (sflag[r * SEQ + j])
                    acc += *(const v4f*)&Wbh[(size_t)stok[r * SEQ + j] * HID + 4 * q];
            }
            *(v4f*)&hsm[r * HS + 4 * q] = lrelu4(acc);   // 16B-aligned (HS % 4 == 0)
        }
        __syncthreads();
        heads_wmma(hsm, Wbp, bbp, Wbe, bbe, bow_e, bow_p, row0);
    } else {
        // ---------------- BEO path: h = lrelu(x @ Weh + beh), fp32 WMMA ----------------
        const int wave = tid >> 5;
        const int lane = tid & 31;
        const int half = lane >> 4;
        const int lr   = lane & 15;
        const int n0   = wave * 16;           // 8 waves cover output cols 0..127 only

        v8f c = {};
        #pragma unroll 4
        for (int kk = 0; kk < BEO; kk += 4) {
            const int ka = kk + half * 2;
            v2f a = *(const v2f*)&x[(size_t)(row0 + lr) * BEO + ka];  // xn == x (MEAN=0,STD=1)
            v2f b;
            b.x = Weh[(size_t)ka * HID + n0 + lr];
            b.y = Weh[(size_t)(ka + 1) * HID + n0 + lr];
            c = __builtin_amdgcn_wmma_f32_16x16x4_f32(false, a, false, b,
                                                      (short)0, c, false, false);
        }
        const float bias = beh[n0 + lr];
        #pragma unroll
        for (int i = 0; i < 8; ++i)
            hsm[(i + 8 * half) * HS + n0 + lr] = lrelu(c[i] + bias);
        __syncthreads();
        heads_wmma(hsm, Wep, bep, Wee, bee, beo_e, beo_p, row0);
    }
}

extern "C" void kernel_launch(void* const* d_in, const int* in_sizes, int n_in,
                              void* d_out, int out_size, void* d_ws, size_t ws_size,
                              hipStream_t stream) {
    const int*   s   = (const int*)  d_in[0];
    const float* x   = (const float*)d_in[1];
    const float* Wbh = (const float*)d_in[2];
    const float* bbh = (const float*)d_in[3];
    const float* Wbp = (const float*)d_in[4];
    const float* bbp = (const float*)d_in[5];
    const float* Wbe = (const float*)d_in[6];
    const float* bbe = (const float*)d_in[7];
    const float* Weh = (const float*)d_in[8];
    const float* beh = (const float*)d_in[9];
    const float* Wep = (const float*)d_in[10];
    const float* bep = (const float*)d_in[11];
    const float* Wee = (const float*)d_in[12];
    const float* bee = (const float*)d_in[13];
    float* out = (float*)d_out;

    dim3 grid(2 * (BATCH / 16));   // 128 BoW blocks + 128 BEO blocks
    bow_beo_kernel<<<grid, 256, 0, stream>>>(s, x, Wbh, bbh, Wbp, bbp, Wbe, bbe,
                                             Weh, beh, Wep, bep, Wee, bee, out);
}